// MMoEModel_36361193128023
// MI455X (gfx1250) — compile-verified
//
#include <hip/hip_runtime.h>
#include <hip/hip_bf16.h>
#include <math.h>
#include <stdint.h>

// ---------------------------------------------------------------------------
// MMoE forward on gfx1250 (CDNA5, wave32):
//   - V_WMMA_F32_16X16X4_F32 for all large GEMMs (exact fp32 path)
//   - Tensor Data Mover (tensor_load_to_lds) for double-buffered LDS staging
//   - weights pre-transposed once per call so A and B fragments are both
//     contiguous v2f LDS loads (no repack movs)
// ---------------------------------------------------------------------------

typedef __attribute__((ext_vector_type(2))) float v2f;
typedef __attribute__((ext_vector_type(8))) float v8f;
typedef __attribute__((ext_vector_type(4))) unsigned int v4u;
typedef __attribute__((ext_vector_type(4))) int v4i;
typedef __attribute__((ext_vector_type(8))) int v8i;

#define BM 128
#define BN 128
#define BK 32
#define LDK (BK + 4)   // 36-float padded rows (TDM pad: 32 DW + 4 DW)

enum { MODE_STORE_RELU = 1, MODE_COMBINE = 2 };

// Issue one TDM load of a [128 rows x 32 dwords] f32 tile from a row-major
// matrix (row stride ldElems) into LDS at byte offset ldsOff, with 4-dword
// padding inserted after every 32 dwords (-> 36-float LDS rows).
// D# layout per CDNA5 ISA ch.10.8 / CDNA5 ISA.
__device__ __forceinline__ void tdm_load_tile_f32(const float* gptr,
                                                  unsigned ldsOff,
                                                  unsigned long long ldElems)
{
    const unsigned long long ga = (unsigned long long)(uintptr_t)gptr;

    v4u g0;
    g0[0] = 1u;                                    // count=1 valid descriptor
    g0[1] = ldsOff;                                // lds_addr (bytes)
    g0[2] = (unsigned)(ga & 0xffffffffu);          // global_addr[31:0]
    g0[3] = (unsigned)((ga >> 32) & 0x01ffffffu)   // global_addr[56:32]
          | 0x80000000u;                           // type=2 ("image") @127:126

    const unsigned tdim0 = BK;     // cols (fastest dim), dwords
    const unsigned tdim1 = BM;     // rows
    const unsigned padInt = 4u;    // 2^(4+1) = 32 dwords before each pad
    const unsigned padAmt = 3u;    // 3+1   = 4 dwords of pad
    v8i g1;
    g1[0] = (int)((2u << 16)          // data_size = 2 -> 4 bytes
                | (1u << 20)          // pad_enable
                | (padInt << 22)      // pad_interval
                | (padAmt << 25));    // pad_amount
    g1[1] = (int)((tdim0 & 0xffffu) << 16);                 // tensor_dim0 lo16
    g1[2] = (int)((tdim0 >> 16) | ((tdim1 & 0xffffu) << 16)); // dim0 hi / dim1 lo
    g1[3] = (int)((tdim1 >> 16) | ((unsigned)BK << 16));    // dim1 hi / tile_dim0
    g1[4] = (int)BM;                                        // tile_dim1 (tile_dim2=0)
    g1[5] = (int)(ldElems & 0xffffffffull);                 // tensor_dim0_stride lo
    g1[6] = (int)((ldElems >> 32) & 0xffffull);             // stride hi (dim1_stride=0)
    g1[7] = 0;

    const v4i z4 = {0, 0, 0, 0};
#if __clang_major__ >= 23
    const v8i z8 = {0, 0, 0, 0, 0, 0, 0, 0};
    __builtin_amdgcn_tensor_load_to_lds(g0, g1, z4, z4, z8, 0);
#else
    __builtin_amdgcn_tensor_load_to_lds(g0, g1, z4, z4, 0);
#endif
}

// C = relu(A * Wt^T + bias)  (Wt is [N][K] row-major, i.e. W pre-transposed).
// MODE_COMBINE additionally multiplies by per-task gates and accumulates into
// the three task_in buffers instead of storing C.
__global__ __launch_bounds__(256) void wmma_gemm_f32(
    const float* __restrict__ A, long long lda, long long Abatch,
    const float* __restrict__ Wt, long long Wbatch,
    const float* __restrict__ bias, long long biasBatch,
    float* __restrict__ C, long long ldc, long long Cbatch,
    int M, int N, int K, int mode,
    const float* __restrict__ gates, float* __restrict__ task_in,
    int expert, int accumulate, int Btot)
{
    __shared__ __align__(16) float As[2][BM][LDK];   // 2 x 18KB
    __shared__ __align__(16) float Bs[2][BN][LDK];   // 2 x 18KB
    __shared__ float gLds[3][BM];

    const int z = blockIdx.z;
    const float* Ab = A + (size_t)z * Abatch;
    const float* Wb = Wt + (size_t)z * Wbatch;
    const float* bb = bias + (size_t)z * biasBatch;
    float* Cb = C + (size_t)z * Cbatch;

    const int rowBase = blockIdx.y * BM;
    const int colBase = blockIdx.x * BN;
    const int tid  = threadIdx.x;
    const int lane = tid & 31;
    const int wave = tid >> 5;      // 0..7
    const int wm   = wave >> 1;     // 0..3 : 32-row slice
    const int wn   = wave & 1;      // 0..1 : 64-col slice
    const int lhalf = lane >> 4;    // K-pair select (ISA 16x4 / 4x16 layouts)
    const int l16   = lane & 15;

    const v8f vzero = {0.f,0.f,0.f,0.f,0.f,0.f,0.f,0.f};
    v8f acc[2][4] = {{vzero,vzero,vzero,vzero},{vzero,vzero,vzero,vzero}};

    const float* Atile = Ab + (size_t)rowBase * lda;   // [BM rows, K cols]
    const float* Btile = Wb + (size_t)colBase * K;     // [BN rows, K cols]
    const int ntiles = K / BK;

    // Prologue: stage tile 0 into buffer 0 (wave0 -> A, wave1 -> B).
    if (wave == 0) tdm_load_tile_f32(Atile, (unsigned)(uintptr_t)&As[0][0][0], (unsigned long long)lda);
    if (wave == 1) tdm_load_tile_f32(Btile, (unsigned)(uintptr_t)&Bs[0][0][0], (unsigned long long)K);

    for (int t = 0; t < ntiles; ++t) {
        const int cur = t & 1;
        if (t + 1 < ntiles) {
            if (wave == 0)
                tdm_load_tile_f32(Atile + (size_t)(t + 1) * BK,
                                  (unsigned)(uintptr_t)&As[cur ^ 1][0][0], (unsigned long long)lda);
            if (wave == 1)
                tdm_load_tile_f32(Btile + (size_t)(t + 1) * BK,
                                  (unsigned)(uintptr_t)&Bs[cur ^ 1][0][0], (unsigned long long)K);
            if (wave < 2) __builtin_amdgcn_s_wait_tensorcnt(1);  // tile t complete (in-order)
        } else {
            if (wave < 2) __builtin_amdgcn_s_wait_tensorcnt(0);
        }
        __syncthreads();   // publish tile t to all waves

        #pragma unroll
        for (int kk = 0; kk < BK; kk += 4) {
            const int ka = kk + lhalf * 2;
            // A 16x4 fragment: M=l16, VGPRs hold K=ka, ka+1 (upper half +2).
            v2f a0 = *(const v2f*)&As[cur][wm * 32 +      l16][ka];
            v2f a1 = *(const v2f*)&As[cur][wm * 32 + 16 + l16][ka];
            // B 4x16 fragment from transposed LDS image: N=row, K contiguous.
            v2f b0 = *(const v2f*)&Bs[cur][wn * 64 +      l16][ka];
            v2f b1 = *(const v2f*)&Bs[cur][wn * 64 + 16 + l16][ka];
            v2f b2 = *(const v2f*)&Bs[cur][wn * 64 + 32 + l16][ka];
            v2f b3 = *(const v2f*)&Bs[cur][wn * 64 + 48 + l16][ka];
            acc[0][0] = __builtin_amdgcn_wmma_f32_16x16x4_f32(false, a0, false, b0, (short)0, acc[0][0], false, false);
            acc[0][1] = __builtin_amdgcn_wmma_f32_16x16x4_f32(false, a0, false, b1, (short)0, acc[0][1], false, false);
            acc[0][2] = __builtin_amdgcn_wmma_f32_16x16x4_f32(false, a0, false, b2, (short)0, acc[0][2], false, false);
            acc[0][3] = __builtin_amdgcn_wmma_f32_16x16x4_f32(false, a0, false, b3, (short)0, acc[0][3], false, false);
            acc[1][0] = __builtin_amdgcn_wmma_f32_16x16x4_f32(false, a1, false, b0, (short)0, acc[1][0], false, false);
            acc[1][1] = __builtin_amdgcn_wmma_f32_16x16x4_f32(false, a1, false, b1, (short)0, acc[1][1], false, false);
            acc[1][2] = __builtin_amdgcn_wmma_f32_16x16x4_f32(false, a1, false, b2, (short)0, acc[1][2], false, false);
            acc[1][3] = __builtin_amdgcn_wmma_f32_16x16x4_f32(false, a1, false, b3, (short)0, acc[1][3], false, false);
        }
        __syncthreads();   // all waves done with tile t before its buffer is re-filled
    }

    if (mode == MODE_COMBINE) {
        if (tid < 3 * BM) {
            const int t = tid / BM, r = tid % BM;
            gLds[t][r] = gates[((size_t)t * Btot + rowBase + r) * 8 + expert];
        }
        __syncthreads();
    }

    // C 16x16 layout: VGPR v -> row v (lanes 0..15) / v+8 (lanes 16..31), col=l16.
    #pragma unroll
    for (int i = 0; i < 2; ++i) {
        #pragma unroll
        for (int j = 0; j < 4; ++j) {
            const int rbase = wm * 32 + i * 16 + lhalf * 8;
            const int gn = colBase + wn * 64 + j * 16 + l16;
            const float bv = bb[gn];
            #pragma unroll
            for (int v = 0; v < 8; ++v) {
                const int r  = rbase + v;
                const int gb = rowBase + r;
                float val = fmaxf(acc[i][j][v] + bv, 0.0f);
                if (mode == MODE_COMBINE) {
                    #pragma unroll
                    for (int t = 0; t < 3; ++t) {
                        const float g = gLds[t][r];
                        float* p = task_in + ((size_t)t * Btot + gb) * (size_t)N + gn;
                        if (accumulate) *p += g * val; else *p = g * val;
                    }
                } else {
                    Cb[(size_t)gb * ldc + gn] = val;
                }
            }
        }
    }
}

// out[z][n][k] = in[z][k][n]  (32x32 LDS tiles, 256 threads)
__global__ __launch_bounds__(256) void transpose_kernel(
    const float* __restrict__ in, float* __restrict__ out, int K, int N)
{
    __shared__ float tile[32][33];
    const int z = blockIdx.z;
    const float* src = in  + (size_t)z * K * N;
    float* dst       = out + (size_t)z * K * N;
    const int k0 = blockIdx.y * 32, n0 = blockIdx.x * 32;
    const int tx = threadIdx.x & 31, ty = threadIdx.x >> 5;
    #pragma unroll
    for (int i = 0; i < 32; i += 8)
        tile[ty + i][tx] = src[(size_t)(k0 + ty + i) * N + (n0 + tx)];
    __syncthreads();
    #pragma unroll
    for (int i = 0; i < 32; i += 8)
        dst[(size_t)(n0 + ty + i) * K + (k0 + tx)] = tile[tx][ty + i];
}

// gates[t][b][e] = softmax_e(x[b,:] @ Wg[t][:,e] + bg[t][e]); E=8 lanes/row.
__global__ __launch_bounds__(256) void gates_kernel(
    const float* __restrict__ x, const float* __restrict__ Wg,
    const float* __restrict__ bg, float* __restrict__ gates, int Btot, int D)
{
    const int tid = blockIdx.x * 256 + threadIdx.x;
    const int b = tid >> 3;
    const int e = tid & 7;
    if (b >= Btot) return;
    const float* xr = x + (size_t)b * D;
    float acc[3] = { bg[0 * 8 + e], bg[1 * 8 + e], bg[2 * 8 + e] };
    for (int d = 0; d < D; ++d) {
        const float xv = xr[d];
        acc[0] = fmaf(xv, Wg[(size_t)(0 * D + d) * 8 + e], acc[0]);
        acc[1] = fmaf(xv, Wg[(size_t)(1 * D + d) * 8 + e], acc[1]);
        acc[2] = fmaf(xv, Wg[(size_t)(2 * D + d) * 8 + e], acc[2]);
    }
    #pragma unroll
    for (int t = 0; t < 3; ++t) {
        float m = acc[t];
        for (int off = 1; off < 8; off <<= 1) m = fmaxf(m, __shfl_xor(m, off, 8));
        const float ex = expf(acc[t] - m);
        float s = ex;
        for (int off = 1; off < 8; off <<= 1) s += __shfl_xor(s, off, 8);
        gates[((size_t)t * Btot + b) * 8 + e] = ex / s;
    }
}

// preds[t*B+b] = sigmoid(th[t][b][:] @ Wt2[t] + bt2[t]); T=256.
__global__ __launch_bounds__(256) void tower2_kernel(
    const float* __restrict__ th, const float* __restrict__ Wt2,
    const float* __restrict__ bt2, float* __restrict__ out, int Btot, int T)
{
    const int i = blockIdx.x * 256 + threadIdx.x;   // i = t*Btot + b
    if (i >= 3 * Btot) return;
    const int t = i / Btot;
    const float* row = th  + (size_t)i * T;
    const float* w   = Wt2 + (size_t)t * T;
    float acc = bt2[t];
    for (int k = 0; k < T; k += 4) {
        const float4 a = *(const float4*)(row + k);
        const float4 b = *(const float4*)(w + k);
        acc = fmaf(a.x, b.x, acc);
        acc = fmaf(a.y, b.y, acc);
        acc = fmaf(a.z, b.z, acc);
        acc = fmaf(a.w, b.w, acc);
    }
    out[i] = 1.0f / (1.0f + expf(-acc));
}

extern "C" void kernel_launch(void* const* d_in, const int* in_sizes, int n_in,
                              void* d_out, int out_size, void* d_ws, size_t ws_size,
                              hipStream_t stream) {
    const int Bt = 16384, D = 1024, E = 8, H = 512, T = 256;

    const float* x   = (const float*)d_in[0];
    const float* We1 = (const float*)d_in[1];
    const float* be1 = (const float*)d_in[2];
    const float* We2 = (const float*)d_in[3];
    const float* be2 = (const float*)d_in[4];
    const float* Wg  = (const float*)d_in[5];
    const float* bg  = (const float*)d_in[6];
    const float* Wt1 = (const float*)d_in[7];
    const float* bt1 = (const float*)d_in[8];
    const float* Wt2 = (const float*)d_in[9];
    const float* bt2 = (const float*)d_in[10];

    // Workspace layout (bytes):
    //   task_in [3,B,H] : 0          .. 100,663,296
    //   th      [3,B,T] : 100,663,296 .. 150,994,944
    //   gates   [3,B,E] : 150,994,944 .. 152,567,808
    //   h1e     [B,H]   : 152,567,808 .. 186,122,240
    //   We1t  [E,H,D]   : 186,122,240 .. 202,899,456
    //   We2t  [E,H,H]   : 202,899,456 .. 211,288,064
    //   Wt1t  [3,T,H]   : 211,288,064 .. 212,860,928
    float* task_in = (float*)d_ws;
    float* thbuf   = (float*)((char*)d_ws + 100663296ull);
    float* gatesB  = (float*)((char*)d_ws + 150994944ull);
    float* h1e     = (float*)((char*)d_ws + 152567808ull);
    float* We1t    = (float*)((char*)d_ws + 186122240ull);
    float* We2t    = (float*)((char*)d_ws + 202899456ull);
    float* Wt1t    = (float*)((char*)d_ws + 211288064ull);

    // 0) Transpose weights so GEMM B-operands are [N][K] row-major.
    { dim3 g(H / 32, D / 32, E); transpose_kernel<<<g, 256, 0, stream>>>(We1, We1t, D, H); }
    { dim3 g(H / 32, H / 32, E); transpose_kernel<<<g, 256, 0, stream>>>(We2, We2t, H, H); }
    { dim3 g(T / 32, H / 32, 3); transpose_kernel<<<g, 256, 0, stream>>>(Wt1, Wt1t, H, T); }

    // 1) Gates + softmax.
    gates_kernel<<<(Bt * 8 + 255) / 256, 256, 0, stream>>>(x, Wg, bg, gatesB, Bt, D);

    // 2) Experts: L1 GEMM -> h1e, then L2 GEMM with fused gate-combine.
    for (int e = 0; e < E; ++e) {
        dim3 g1(H / BN, Bt / BM, 1);
        wmma_gemm_f32<<<g1, 256, 0, stream>>>(
            x, D, 0,
            We1t + (size_t)e * D * H, 0,
            be1 + (size_t)e * H, 0,
            h1e, H, 0,
            Bt, H, D, MODE_STORE_RELU,
            nullptr, nullptr, 0, 0, Bt);

        dim3 g2(H / BN, Bt / BM, 1);
        wmma_gemm_f32<<<g2, 256, 0, stream>>>(
            h1e, H, 0,
            We2t + (size_t)e * H * H, 0,
            be2 + (size_t)e * H, 0,
            nullptr, 0, 0,
            Bt, H, H, MODE_COMBINE,
            gatesB, task_in, e, (e > 0) ? 1 : 0, Bt);
    }

    // 3) Towers layer 1 (3 tasks via grid.z).
    dim3 g3(T / BN, Bt / BM, 3);
    wmma_gemm_f32<<<g3, 256, 0, stream>>>(
        task_in, H, (long long)Bt * H,
        Wt1t, (long long)H * T,
        bt1, T,
        thbuf, T, (long long)Bt * T,
        Bt, T, H, MODE_STORE_RELU,
        nullptr, nullptr, 0, 0, Bt);

    // 4) Tower layer 2 + sigmoid -> d_out (preds concat t*B+b).
    tower2_kernel<<<(3 * Bt + 255) / 256, 256, 0, stream>>>(
        thbuf, Wt2, bt2, (float*)d_out, Bt, T);
}